// RoIAlign_51805895525040
// MI455X (gfx1250) — compile-verified
//
#include <hip/hip_runtime.h>

typedef __attribute__((ext_vector_type(2))) float v2f;
typedef __attribute__((ext_vector_type(8))) float v8f;

#define PHB 7
#define PWB 7
#define SCALE_F 0.0625f
#define CDIM 256
#define HDIM 100
#define WDIM 152
#define SPAN 40   // max per-axis patch span is 32; padded

struct __align__(32) RoiShared {
    float Ax[16][SPAN];   // x-weights; rows 7..15 stay zero
    float AyT[SPAN][8];   // transposed y-weights: AyT[row][ph], col 7 zero
    int x0, y0, chunksx, rows, bidx;
};

__device__ __forceinline__ v8f v8zero() {
    return (v8f){0.f, 0.f, 0.f, 0.f, 0.f, 0.f, 0.f, 0.f};
}

__device__ __forceinline__ v8f wmma_f32(v2f a, v2f b, v8f c) {
    return __builtin_amdgcn_wmma_f32_16x16x4_f32(false, a, false, b, (short)0, c,
                                                 false, false);
}

// One channel-block (16 channels) of one ROI. CH = compile-time chunk count.
// A rows are duplicated (rows 8..14 == rows 0..6), so each accumulator pair
// holds ph=2p in lanes 0..15 and ph=2p+1 in lanes 16..31.
template <int CH>
__device__ __forceinline__ void roi_cb_run(
    const float* __restrict__ chan, int rows, int half,
    const v2f* __restrict__ av, const int* __restrict__ coff,
    const RoiShared& sh, float* __restrict__ op)
{
    v8f acc[4];
    #pragma unroll
    for (int p = 0; p < 4; ++p) acc[p] = v8zero();

    v2f bufA[CH], bufB[CH];

    auto loadrow = [&](const float* __restrict__ rowp, v2f* __restrict__ buf) {
        #pragma unroll
        for (int k = 0; k < CH; ++k) {
            const float* ptr = rowp + coff[k];
            buf[k].x = ptr[0];
            buf[k].y = ptr[1];
        }
    };
    auto compute = [&](const v2f* __restrict__ buf, int r) {
        v8f G = v8zero();
        #pragma unroll
        for (int k = 0; k < CH; ++k) G = wmma_f32(av[k], buf[k], G);
        float ayv[4];
        #pragma unroll
        for (int p = 0; p < 4; ++p) ayv[p] = sh.AyT[r][2 * p + half];
        #pragma unroll
        for (int p = 0; p < 4; ++p) acc[p] += G * ayv[p];
    };

    // software pipeline: row r+1 loads in flight while row r computes
    loadrow(chan, bufA);
    int r = 0;
    for (;;) {
        if (r + 1 < rows) loadrow(chan + (size_t)(r + 1) * WDIM, bufB);
        compute(bufA, r);
        if (++r >= rows) break;
        if (r + 1 < rows) loadrow(chan + (size_t)(r + 1) * WDIM, bufA);
        compute(bufB, r);
        if (++r >= rows) break;
    }

    // lanes 0..15: ph = 0,2,4,6 ; lanes 16..31: ph = 1,3,5 (p=3 -> ph=7 skipped)
    #pragma unroll
    for (int p = 0; p < 4; ++p) {
        const int ph = 2 * p + half;
        if (ph < PHB) {
            #pragma unroll
            for (int v = 0; v < PWB; ++v)
                op[ph * PWB + v] = acc[p][v];
        }
    }
}

__global__ __launch_bounds__(256) void roialign_wmma_kernel(
    const float* __restrict__ feat,
    const float* __restrict__ rois,
    float* __restrict__ out,
    int N)
{
    __shared__ RoiShared sh;
    const int n   = blockIdx.x;
    const int tid = threadIdx.x;

    // zero Ax + AyT (contiguous: 16*SPAN + SPAN*8 floats)
    {
        float* z = &sh.Ax[0][0];
        for (int i = tid; i < 16 * SPAN + SPAN * 8; i += 256) z[i] = 0.0f;
    }
    __syncthreads();

    if (tid < 2) {
        const bool isx = (tid == 0);
        const float size = isx ? (float)WDIM : (float)HDIM;
        const float r1 = rois[(size_t)n * 5 + (isx ? 1 : 2)] * SCALE_F;
        const float r2 = rois[(size_t)n * 5 + (isx ? 3 : 4)] * SCALE_F;
        const float roi = fmaxf(r2 - r1, 1.0f);
        const float bin = roi / 7.0f;
        int   lo[14];
        float wlo[14], whi[14];
        for (int j = 0; j < 14; ++j) {
            const int p = j >> 1, s = j & 1;
            float c = r1 + (float)p * bin + ((float)s + 0.5f) * bin * 0.5f;
            const bool valid = (c > -1.0f) && (c < size);
            c = fminf(fmaxf(c, 0.0f), size - 1.0f);
            const float lof = fminf(floorf(c), size - 2.0f);
            lo[j] = (int)lof;
            const float f = c - lof;
            wlo[j] = valid ? 0.5f * (1.0f - f) : 0.0f;  // 0.5 folds SR-mean per axis
            whi[j] = valid ? 0.5f * f          : 0.0f;
        }
        const int o0 = lo[0];  // sample coords monotonic -> lo[0] is min
        for (int j = 0; j < 14; ++j) {
            const int bi = j >> 1;
            const int d  = lo[j] - o0;
            if (isx) {
                sh.Ax[bi][d]     += wlo[j];
                sh.Ax[bi][d + 1] += whi[j];
            } else {
                sh.AyT[d][bi]     += wlo[j];
                sh.AyT[d + 1][bi] += whi[j];
            }
        }
        const int span = lo[13] - o0 + 2;
        if (isx) { sh.x0 = o0; sh.chunksx = (span + 3) >> 2; }
        else     { sh.y0 = o0; sh.rows    = span; }
    }
    if (tid == 2) sh.bidx = (int)rois[(size_t)n * 5 + 0];
    __syncthreads();

    const int lane = tid & 31;
    const int wave = tid >> 5;
    const int m    = lane & 15;   // A-row / B-col (channel) index
    const int half = lane >> 4;   // K half-select; also selects odd/even ph
    const int x0 = sh.x0, y0 = sh.y0, chunks = sh.chunksx, rows = sh.rows;
    const int b  = sh.bidx;

    // loop-invariant per wave: A-fragments (rows duplicated via m&7) + clamped cols
    const int ma = m & 7;         // rows 8..14 replicate rows 0..6; 7/15 -> zero row
    v2f av[8];
    int coff[8];
    #pragma unroll
    for (int k = 0; k < 8; ++k) {
        const int xoff = k * 4 + 2 * half;
        av[k].x = sh.Ax[ma][xoff];
        av[k].y = sh.Ax[ma][xoff + 1];
        int col = x0 + xoff;
        coff[k] = (col < WDIM - 2) ? col : (WDIM - 2);  // clamp: weights there are 0
    }

    // each wave handles channel-blocks {wave, wave+8}
    for (int cb = wave; cb < 16; cb += 8) {
        const int c = cb * 16 + m;
        const float* chan =
            feat + (((size_t)b * CDIM + c) * HDIM + y0) * (size_t)WDIM;
        float* op = out + ((size_t)n * CDIM + c) * (PHB * PWB);

        if (chunks <= 2)
            roi_cb_run<2>(chan, rows, half, av, coff, sh, op);
        else if (chunks <= 4)
            roi_cb_run<4>(chan, rows, half, av, coff, sh, op);
        else
            roi_cb_run<8>(chan, rows, half, av, coff, sh, op);
    }
}

extern "C" void kernel_launch(void* const* d_in, const int* in_sizes, int n_in,
                              void* d_out, int out_size, void* d_ws, size_t ws_size,
                              hipStream_t stream)
{
    const float* feat = (const float*)d_in[0];
    const float* rois = (const float*)d_in[1];
    float* out        = (float*)d_out;
    const int N = in_sizes[1] / 5;   // 1000 ROIs
    roialign_wmma_kernel<<<N, 256, 0, stream>>>(feat, rois, out, N);
}